// Encoder_16415365005698
// MI455X (gfx1250) — compile-verified
//
#include <hip/hip_runtime.h>

typedef __attribute__((ext_vector_type(2))) float v2f;
typedef __attribute__((ext_vector_type(8))) float v8f;

#define TPB 256

// ---------------- degree / norm precompute ----------------

__global__ void deg_init_kernel(float* deg, int n) {
  int i = blockIdx.x * blockDim.x + threadIdx.x;
  if (i < n) deg[i] = 1.0f;  // self-loop contributes 1
}

__global__ void deg_count_kernel(const int* __restrict__ col, float* deg, int E) {
  int e = blockIdx.x * blockDim.x + threadIdx.x;
  if (e < E) atomicAdd(&deg[col[e]], 1.0f);
}

__global__ void dinv_kernel(float* deg, int n) {
  int i = blockIdx.x * blockDim.x + threadIdx.x;
  if (i < n) deg[i] = rsqrtf(deg[i]);  // deg >= 1 always (self-loops)
}

__global__ void norm_kernel(const int* __restrict__ row, const int* __restrict__ col,
                            const float* __restrict__ dinv, float* __restrict__ norm, int E) {
  int e = blockIdx.x * blockDim.x + threadIdx.x;
  if (e < E) norm[e] = dinv[row[e]] * dinv[col[e]];
}

// ---------------- dense transform + fused self-loop/bias epilogue ----------------
// T[N,F] = act(X[N,K]) @ W[K,F]          (raw transform, gathered by edge scatter)
// Y[N,F] = T * dinv^2 + bias             (aggregation accumulator init)
// One wave computes one 16x16 output tile with V_WMMA_F32_16X16X4_F32.
// A (16x4 f32): lanes 0-15 hold M=0..15 K={k0,k0+1}; lanes 16-31 K={k0+2,k0+3}.
// B (4x16 f32): mirrored striping. C/D: VGPR v holds M=v+half*8, N=lane&15.
template <int K, int F, bool RELU>
__global__ void gcn_gemm_kernel(const float* __restrict__ X, const float* __restrict__ Wm,
                                const float* __restrict__ bias, const float* __restrict__ dinv,
                                float* __restrict__ T, float* __restrict__ Y, int N) {
  const int lane = threadIdx.x & 31;
  const int wave = threadIdx.x >> 5;
  const int m0 = (blockIdx.x * 4 + wave) * 16;
  const int n0 = blockIdx.y * 16;
  if (m0 >= N) return;                  // uniform per wave: safe around WMMA
  const int half = lane >> 4;           // selects K pair
  const int lm   = lane & 15;           // M (A) / N (B, C)
  int rA = m0 + lm;
  if (rA >= N) rA = N - 1;              // clamp so EXEC stays all-1s
  const float* __restrict__ xr = X + (size_t)rA * K + half * 2;
  const float* __restrict__ wc = Wm + (size_t)(half * 2) * F + n0 + lm;
  v8f c = {};
#pragma unroll
  for (int k0 = 0; k0 < K; k0 += 4) {
    v2f a = *(const v2f*)(xr + k0);     // 8B-aligned: K % 4 == 0
    if (RELU) { a.x = fmaxf(a.x, 0.0f); a.y = fmaxf(a.y, 0.0f); }
    v2f b;
    b.x = wc[(size_t)k0 * F];
    b.y = wc[(size_t)(k0 + 1) * F];
    c = __builtin_amdgcn_wmma_f32_16x16x4_f32(false, a, false, b, (short)0, c, false, false);
  }
  const float bn = bias[n0 + lm];
#pragma unroll
  for (int v = 0; v < 8; ++v) {
    int m = m0 + half * 8 + v;
    if (m < N) {
      float s = dinv[m];
      size_t o = (size_t)m * F + n0 + lm;
      T[o] = c[v];
      Y[o] = c[v] * (s * s) + bn;
    }
  }
}

// ---------------- edge scatter-add (memory-bound hot loop) ----------------

template <int F>
__global__ void edge_scatter_kernel(const float* __restrict__ t, const int* __restrict__ row,
                                    const int* __restrict__ col, const float* __restrict__ norm,
                                    float* __restrict__ y, int E) {
  constexpr int FQ = F >> 2;            // power of two -> div/mod become shifts
  long long idx = (long long)blockIdx.x * blockDim.x + threadIdx.x;
  if (idx >= (long long)E * FQ) return;
  int e = (int)(idx / FQ);
  int q = (int)(idx - (long long)e * FQ);
  float w = norm[e];
  int r = row[e];
  int c = col[e];
  float4 v = ((const float4*)t)[(size_t)r * FQ + q];
  float* yp = y + (size_t)c * F + (q << 2);
  atomicAdd(yp + 0, v.x * w);
  atomicAdd(yp + 1, v.y * w);
  atomicAdd(yp + 2, v.z * w);
  atomicAdd(yp + 3, v.w * w);
}

// ---------------- per-layer driver ----------------

template <int K, int F, bool RELU>
static void run_layer(const float* in, const float* Wm, const float* bias,
                      const float* dinv, const float* norm,
                      const int* erow, const int* ecol,
                      float* tbuf, float* ybuf, int N, int E, hipStream_t stream) {
  dim3 g((unsigned)((N + 63) / 64), (unsigned)(F / 16));
  gcn_gemm_kernel<K, F, RELU><<<g, 128, 0, stream>>>(in, Wm, bias, dinv, tbuf, ybuf, N);
  long long eq = (long long)E * (F / 4);
  unsigned gb = (unsigned)((eq + TPB - 1) / TPB);
  edge_scatter_kernel<F><<<gb, TPB, 0, stream>>>(tbuf, erow, ecol, norm, ybuf, E);
}

// ---------------- host-side orchestration ----------------

extern "C" void kernel_launch(void* const* d_in, const int* in_sizes, int n_in,
                              void* d_out, int out_size, void* d_ws, size_t ws_size,
                              hipStream_t stream) {
  const float* x  = (const float*)d_in[0];
  const int*   ei = (const int*)d_in[1];
  const float* W1 = (const float*)d_in[2];  const float* b1 = (const float*)d_in[3];
  const float* W2 = (const float*)d_in[4];  const float* b2 = (const float*)d_in[5];
  const float* W3 = (const float*)d_in[6];  const float* b3 = (const float*)d_in[7];
  const float* W4 = (const float*)d_in[8];  const float* b4 = (const float*)d_in[9];
  const float* Wmu = (const float*)d_in[10]; const float* bmu = (const float*)d_in[11];
  const float* Wls = (const float*)d_in[12]; const float* bls = (const float*)d_in[13];

  const int N = in_sizes[0] / 128;  // 50000
  const int E = in_sizes[1] / 2;    // 800000
  const int* erow = ei;
  const int* ecol = ei + E;

  float* ws = (float*)d_ws;
  size_t off = 0;
  float* dinv = ws + off; off += (size_t)N;
  float* norm = ws + off; off += (size_t)E;
  float* tbuf = ws + off; off += (size_t)N * 256;
  float* hA   = ws + off; off += (size_t)N * 256;
  float* hB   = ws + off;

  deg_init_kernel<<<(N + TPB - 1) / TPB, TPB, 0, stream>>>(dinv, N);
  deg_count_kernel<<<(E + TPB - 1) / TPB, TPB, 0, stream>>>(ecol, dinv, E);
  dinv_kernel<<<(N + TPB - 1) / TPB, TPB, 0, stream>>>(dinv, N);
  norm_kernel<<<(E + TPB - 1) / TPB, TPB, 0, stream>>>(erow, ecol, dinv, norm, E);

  float* out_mu = (float*)d_out;
  float* out_ls = (float*)d_out + (size_t)N * 16;

  // 128 -> 256 -> 128 -> 64 -> 32 -> {16, 16}; ReLU fused into next layer's A load
  run_layer<128, 256, false>(x,  W1,  b1,  dinv, norm, erow, ecol, tbuf, hA,     N, E, stream);
  run_layer<256, 128, true >(hA, W2,  b2,  dinv, norm, erow, ecol, tbuf, hB,     N, E, stream);
  run_layer<128,  64, true >(hB, W3,  b3,  dinv, norm, erow, ecol, tbuf, hA,     N, E, stream);
  run_layer< 64,  32, true >(hA, W4,  b4,  dinv, norm, erow, ecol, tbuf, hB,     N, E, stream);
  run_layer< 32,  16, true >(hB, Wmu, bmu, dinv, norm, erow, ecol, tbuf, out_mu, N, E, stream);
  run_layer< 32,  16, true >(hB, Wls, bls, dinv, norm, erow, ecol, tbuf, out_ls, N, E, stream);
}